// SoftResampler_8864812499227
// MI455X (gfx1250) — compile-verified
//
#include <hip/hip_runtime.h>
#include <hip/hip_bf16.h>

// Problem constants (from reference): particles [B,K,H] f32, log_weights [B,K] f32
#define BB 1024
#define KK 128
#define HH 512
#define ALPHA 0.5f
#define EPSV 1e-10f

typedef unsigned int u32x4 __attribute__((ext_vector_type(4)));
typedef int          i32x4 __attribute__((ext_vector_type(4)));
typedef int          i32x8 __attribute__((ext_vector_type(8)));

// ---------------------------------------------------------------------------
// Kernel 1: proposal + inverse-CDF multinomial sampling + new log-weights.
// One block per batch row, one thread per particle slot. Tiny vs the gather.
// ---------------------------------------------------------------------------
__device__ __forceinline__ float hash_u01(unsigned long long x) {
    // splitmix64-style mix -> uniform in [0,1)
    x += 0x9E3779B97F4A7C15ULL;
    x ^= x >> 33; x *= 0xFF51AFD7ED558CCDULL;
    x ^= x >> 33; x *= 0xC4CEB9FE1A85EC53ULL;
    x ^= x >> 33;
    return (float)(unsigned)(x >> 40) * (1.0f / 16777216.0f); // 24-bit mantissa
}

__global__ void soft_resample_weights_kernel(const float* __restrict__ log_w,
                                             float* __restrict__ out_lw,
                                             int* __restrict__ anc) {
    __shared__ float prop_s[KK];
    __shared__ float lw_s[KK];
    __shared__ float cdf[KK];
    __shared__ float nl_s[KK];

    const int b = blockIdx.x;
    const int t = threadIdx.x;

    const float lw = log_w[b * KK + t];
    const float p  = ALPHA * __expf(lw) + (1.0f - ALPHA) / (float)KK;
    prop_s[t] = p;
    lw_s[t]   = lw;
    cdf[t]    = p;
    __syncthreads();

    // Inclusive prefix sum (Hillis-Steele) over K=128 in LDS.
    #pragma unroll
    for (int off = 1; off < KK; off <<= 1) {
        float add = (t >= off) ? cdf[t - off] : 0.0f;
        __syncthreads();
        cdf[t] += add;
        __syncthreads();
    }
    const float total = cdf[KK - 1];

    // Draw uniform, inverse-CDF binary search -> ancestor index.
    const unsigned long long ctr =
        (unsigned long long)(42u) * 0x100000000ULL + (unsigned long long)(b * KK + t);
    const float target = hash_u01(ctr) * total;
    int lo = 0, hi = KK - 1;
    while (lo < hi) {
        int mid = (lo + hi) >> 1;
        if (cdf[mid] < target) lo = mid + 1; else hi = mid;
    }
    const int a = lo;
    anc[b * KK + t] = a;

    // Importance-weight correction: lw_g - log(prop_g + eps), then logsumexp norm.
    const float nl = lw_s[a] - __logf(prop_s[a] + EPSV);
    nl_s[t] = nl;
    __syncthreads();

    float m = -3.4e38f;
    #pragma unroll 8
    for (int j = 0; j < KK; ++j) m = fmaxf(m, nl_s[j]);
    float s = 0.0f;
    #pragma unroll 8
    for (int j = 0; j < KK; ++j) s += __expf(nl_s[j] - m);

    out_lw[b * KK + t] = nl - (m + __logf(s));
}

// ---------------------------------------------------------------------------
// Kernel 2: pure-TDM gather + writeback. One wave per 16 output rows.
//   tensor_load_to_lds  : gather-mode D#, 16 ancestor rows (2KB each) -> LDS
//   tensor_store_from_lds: normal-mode D#, contiguous 32KB LDS -> output
// The DMA engines move all 512MB; the wave only builds descriptors.
// ---------------------------------------------------------------------------
__device__ __forceinline__ void tdm_load(u32x4 g0, i32x8 g1, i32x4 g2, i32x4 g3) {
#if defined(__clang_major__) && (__clang_major__ >= 23)
    i32x8 gx = {};
    __builtin_amdgcn_tensor_load_to_lds(g0, g1, g2, g3, gx, 0);
#else
    __builtin_amdgcn_tensor_load_to_lds(g0, g1, g2, g3, 0);
#endif
}

__device__ __forceinline__ void tdm_store(u32x4 g0, i32x8 g1, i32x4 g2, i32x4 g3) {
#if defined(__clang_major__) && (__clang_major__ >= 23)
    i32x8 gx = {};
    __builtin_amdgcn_tensor_store_from_lds(g0, g1, g2, g3, gx, 0);
#else
    __builtin_amdgcn_tensor_store_from_lds(g0, g1, g2, g3, 0);
#endif
}

__global__ void tdm_gather_kernel(const float* __restrict__ particles,
                                  const int* __restrict__ anc,
                                  float* __restrict__ out) {
    __shared__ float4 stage4[2048];   // 16 rows x 512 f32 = 32 KB

    const int b  = blockIdx.x >> 3;        // batch
    const int g  = blockIdx.x & 7;         // row-group within batch
    const int k0 = g * 16;

    // Ancestor indices for this group (wave-uniform scalar values).
    const int* ap = anc + b * KK + k0;
    int idx[16];
    #pragma unroll
    for (int i = 0; i < 16; ++i) idx[i] = __builtin_amdgcn_readfirstlane(ap[i]);

    const unsigned long long ga =
        (unsigned long long)(uintptr_t)(particles + (size_t)b * KK * HH);
    const unsigned long long gd =
        (unsigned long long)(uintptr_t)(out + ((size_t)(b * KK + k0)) * HH);
    const unsigned lds_addr = (unsigned)(uintptr_t)(&stage4[0]);

    // ======== Load descriptor: gather mode, 16-bit row indices ========
    u32x4 L0;
    L0.x = 0x80000001u;                                       // gather_mode | count=1
    L0.y = lds_addr;                                          // lds_addr[31:0]
    L0.z = (unsigned)(ga & 0xFFFFFFFFu);                      // global_addr[31:0]
    L0.w = (unsigned)((ga >> 32) & 0x01FFFFFFu) | (2u << 30); // addr[56:32] | type=2

    i32x8 L1;
    L1[0] = 0x00020000;                    // data_size=2 (4 bytes)
    L1[1] = (int)(((unsigned)HH) << 16);   // tensor_dim0 = 512
    L1[2] = (int)(((unsigned)KK) << 16);   // tensor_dim1 = 128
    L1[3] = (int)(((unsigned)HH) << 16);   // tile_dim0 = 512 (full row)
    L1[4] = 16;                            // tile_dim1 = 16 valid indices
    L1[5] = HH;                            // tensor_dim0_stride = 512
    L1[6] = 0;
    L1[7] = 0;

    i32x4 L2, L3;                          // sixteen 16-bit row indices
    #pragma unroll
    for (int j = 0; j < 4; ++j) {
        L2[j] = (idx[2 * j] & 0xFFFF) | (idx[2 * j + 1] << 16);
        L3[j] = (idx[8 + 2 * j] & 0xFFFF) | (idx[9 + 2 * j] << 16);
    }

    // ======== Store descriptor: normal mode, 1-D tile of 8192 f32 ========
    u32x4 S0;
    S0.x = 0x00000001u;                                       // count=1, normal mode
    S0.y = lds_addr;                                          // lds_addr[31:0]
    S0.z = (unsigned)(gd & 0xFFFFFFFFu);                      // global_addr[31:0]
    S0.w = (unsigned)((gd >> 32) & 0x01FFFFFFu) | (2u << 30); // addr[56:32] | type=2

    i32x8 S1;
    S1[0] = 0x00020000;                     // data_size=2 (4 bytes)
    S1[1] = (int)(8192u << 16);             // tensor_dim0 = 8192
    S1[2] = (int)(1u << 16);                // tensor_dim1 = 1
    S1[3] = (int)(8192u << 16);             // tile_dim0 = 8192
    S1[4] = 1;                              // tile_dim1 = 1
    S1[5] = 8192;                           // tensor_dim0_stride = 8192
    S1[6] = 0;
    S1[7] = 0;

    i32x4 Z = {};                           // groups 2/3 unused (dims <= 2D)

    tdm_load(L0, L1, L2, L3);
    __builtin_amdgcn_s_wait_tensorcnt(0);   // gather complete -> LDS valid
    tdm_store(S0, S1, Z, Z);
    __builtin_amdgcn_s_wait_tensorcnt(0);   // writeback complete before endpgm
}

extern "C" void kernel_launch(void* const* d_in, const int* in_sizes, int n_in,
                              void* d_out, int out_size, void* d_ws, size_t ws_size,
                              hipStream_t stream) {
    (void)in_sizes; (void)n_in; (void)out_size; (void)ws_size;

    const float* particles = (const float*)d_in[0];   // [B,K,H] f32
    const float* log_w     = (const float*)d_in[1];   // [B,K]   f32
    float* out    = (float*)d_out;                    // [B,K,H] f32 ...
    float* out_lw = out + (size_t)BB * KK * HH;       // ... then [B,K] f32
    int*   anc    = (int*)d_ws;                       // [B,K] ancestors (scratch)

    soft_resample_weights_kernel<<<BB, KK, 0, stream>>>(log_w, out_lw, anc);
    tdm_gather_kernel<<<BB * (KK / 16), 32, 0, stream>>>(particles, anc, out);
}